// EntangledDeltaTreeModel_11673721110718
// MI455X (gfx1250) — compile-verified
//
#include <hip/hip_runtime.h>
#include <math.h>

typedef __attribute__((ext_vector_type(2))) float v2f;
typedef __attribute__((ext_vector_type(8))) float v8f;

// Tree offsets: OFFS[l] = start of level l, sizes 8^l
#define OFF0 0
#define OFF1 1
#define OFF2 9
#define OFF3 73
#define OFF4 585
#define OFF5 4681
#define OFF6 37449
#define OFF7 299593
#define NNODES  2396745
#define NLEAVES 2097152
#define MIN_DIST 1e-7f

// Accumulate the 8-level path sum for 16 consecutive leaves (leaves 16t..16t+15)
// using one f32 WMMA: D = ones(16x4) x B(4x16) + C.
//   C[m][n]  = leafDelta[m][n] + lvl6parent(m)[n] + anc5[n] + root[n]
//   B rows   = {anc4, anc3, anc2, anc1}[n]   (A all-ones -> K-permutation invariant)
__device__ __forceinline__ void subtile16(const float* __restrict__ deltas, int t, int lane,
                                          float* __restrict__ wbase, float* __restrict__ abase) {
  const int dim  = lane & 15;   // N = lane % 16
  const int half = lane >> 4;   // C/D: lanes 16-31 hold M = j + 8

  // shared ancestors folded directly into C (uniform per row / per half-wave)
  float sh = deltas[(size_t)(OFF6 + 2 * t + half) * 16 + dim]   // level-6 parent of rows in this half
           + deltas[(size_t)(OFF5 + (t >> 2)) * 16 + dim]       // level-5 ancestor
           + deltas[dim];                                       // root (level 0)

  // B: four ancestor rows (levels 4,3,2,1); each half-wave loads 2 contiguous 64B rows
  const int colA = half ? (OFF2 + (t >> 11)) : (OFF4 + (t >> 5));
  const int colB = half ? (OFF1 + (t >> 14)) : (OFF3 + (t >> 8));
  v2f b; b.x = deltas[(size_t)colA * 16 + dim];
         b.y = deltas[(size_t)colB * 16 + dim];
  v2f a; a.x = 1.0f; a.y = 1.0f;   // all-ones incidence

  v8f c;
  const size_t leafrow = (size_t)(OFF7 + 16 * t);
#pragma unroll
  for (int j = 0; j < 8; ++j) {
    const int leaf = j + 8 * half;                    // M index, doc'd C/D layout
    float ld = deltas[(leafrow + leaf) * 16 + dim];   // contiguous 64B per half-wave
    abase[leaf * 17 + dim] = __builtin_fabsf(ld);     // stash |leaf delta| for loss term
    c[j] = ld + sh;
  }

  // D = A*B + C  (exact f32 accumulation)
  v8f d = __builtin_amdgcn_wmma_f32_16x16x4_f32(false, a, false, b, (short)0, c, false, false);

#pragma unroll
  for (int j = 0; j < 8; ++j)
    wbase[(j + 8 * half) * 17 + dim] = d[j];          // stride-17: conflict-light
}

__global__ __launch_bounds__(256) void leaf_kernel(const float* __restrict__ x,
                                                   const float* __restrict__ deltas,
                                                   const float* __restrict__ heights,
                                                   float* __restrict__ out) {
  __shared__ float lds_w[8 * 32 * 17];
  __shared__ float lds_a[8 * 32 * 17];
  const int lane = threadIdx.x & 31;
  const int wib  = threadIdx.x >> 5;                 // wave in block (8 waves)
  const int gw   = blockIdx.x * 8 + wib;             // global wave -> 32 leaves
  float* wbase = &lds_w[wib * 32 * 17];
  float* abase = &lds_a[wib * 32 * 17];

  subtile16(deltas, 2 * gw,     lane, wbase,           abase);
  subtile16(deltas, 2 * gw + 1, lane, wbase + 16 * 17, abase + 16 * 17);

  // One lane per leaf: per-leaf MLP + leaf-level loss term
  const int leaf = gw * 32 + lane;
  const float* wl = &wbase[lane * 17];
  const float x0 = x[(size_t)leaf * 3 + 0];
  const float x1 = x[(size_t)leaf * 3 + 1];
  const float x2 = x[(size_t)leaf * 3 + 2];
  float acc = wl[15];                                  // ob
#pragma unroll
  for (int h = 0; h < 3; ++h) {
    float pre = x0 * wl[h] + x1 * wl[3 + h] + x2 * wl[6 + h] + wl[9 + h];
    acc += tanhf(pre) * wl[12 + h];                    // hidden * ow
  }
  out[leaf] = acc;

  float s = 0.f;
  const float* al = &abase[lane * 17];
#pragma unroll
  for (int i = 0; i < 16; ++i) s += al[i];
  const float hl = heights[OFF7 + leaf];
  const float hp = heights[OFF6 + (leaf >> 3)];
  float term = s / fmaxf(fabsf(hl - hp), MIN_DIST);
#pragma unroll
  for (int off = 16; off > 0; off >>= 1) term += __shfl_xor(term, off, 32);
  if (lane == 0) atomicAdd(out + NLEAVES, term);
}

// delta_loss terms for internal levels 1..6 (nodes 1 .. OFF7-1)
__global__ __launch_bounds__(256) void inner_kernel(const float* __restrict__ deltas,
                                                    const float* __restrict__ heights,
                                                    float* __restrict__ out) {
  const int n = blockIdx.x * blockDim.x + threadIdx.x + 1;
  float term = 0.f;
  if (n < OFF7) {
    int base, pbase;
    if      (n < OFF2) { base = OFF1; pbase = OFF0; }
    else if (n < OFF3) { base = OFF2; pbase = OFF1; }
    else if (n < OFF4) { base = OFF3; pbase = OFF2; }
    else if (n < OFF5) { base = OFF4; pbase = OFF3; }
    else if (n < OFF6) { base = OFF5; pbase = OFF4; }
    else               { base = OFF6; pbase = OFF5; }
    const int parent = pbase + ((n - base) >> 3);
    const float4* dp = (const float4*)(deltas + (size_t)n * 16);
    float s = 0.f;
#pragma unroll
    for (int k = 0; k < 4; ++k) {
      float4 v = dp[k];
      s += __builtin_fabsf(v.x) + __builtin_fabsf(v.y) +
           __builtin_fabsf(v.z) + __builtin_fabsf(v.w);
    }
    term = s / fmaxf(fabsf(heights[n] - heights[parent]), MIN_DIST);
  }
#pragma unroll
  for (int off = 16; off > 0; off >>= 1) term += __shfl_xor(term, off, 32);
  if ((threadIdx.x & 31) == 0 && term != 0.f) atomicAdd(out + NLEAVES, term);
}

__global__ void zero_kernel(float* __restrict__ out) { out[NLEAVES] = 0.f; }

extern "C" void kernel_launch(void* const* d_in, const int* in_sizes, int n_in,
                              void* d_out, int out_size, void* d_ws, size_t ws_size,
                              hipStream_t stream) {
  const float* x       = (const float*)d_in[0];   // (2097152, 3)
  const float* deltas  = (const float*)d_in[1];   // (2396745, 16)
  const float* heights = (const float*)d_in[2];   // (2396745,)
  float* out = (float*)d_out;                     // 2097152 outs + 1 loss

  zero_kernel<<<1, 1, 0, stream>>>(out);

  const int innerThreads = OFF7 - 1;              // 299592 nodes, levels 1..6
  inner_kernel<<<(innerThreads + 255) / 256, 256, 0, stream>>>(deltas, heights, out);

  // 8192 blocks * 8 waves * 32 leaves = 2,097,152 leaves exactly (no divergence -> EXEC all-1s for WMMA)
  leaf_kernel<<<NLEAVES / 256, 256, 0, stream>>>(x, deltas, heights, out);
}